// FinalModel_JIT_8924942041923
// MI455X (gfx1250) — compile-verified
//
#include <hip/hip_runtime.h>
#include <math.h>

// CDNA5 / gfx1250, wave32. Matrix path: V_WMMA_F32_16X16X4_F32 (full fp32,
// M=16 points, N=16 neurons, K in chunks of 4). GELU uses the branchless
// Abramowitz-Stegun 7.1.26 erf (max abs err 1.5e-7, 2 TRANS ops, no EXEC
// divergence anywhere -> WMMA's EXEC-all-ones requirement holds globally).

typedef __attribute__((ext_vector_type(2))) float v2f;
typedef __attribute__((ext_vector_type(8))) float v8f;

#define FM_H 1024
#define FM_W 1024
#define FM_HW (FM_H * FM_W)

// erf(x) = sign(x) * (1 - poly(t)*exp(-x^2)),  t = 1/(1 + 0.3275911*|x|)
// Abramowitz & Stegun 7.1.26; |error| <= 1.5e-7; branchless.
__device__ __forceinline__ float erf_fast(float x) {
    const float NLOG2E = -1.4426950408889634f;  // -log2(e)
    float ax = __builtin_fabsf(x);
    float t  = __builtin_amdgcn_rcpf(__builtin_fmaf(0.3275911f, ax, 1.0f));
    float p  = __builtin_fmaf(1.061405429f, t, -1.453152027f);
    p = __builtin_fmaf(p, t, 1.421413741f);
    p = __builtin_fmaf(p, t, -0.284496736f);
    p = __builtin_fmaf(p, t, 0.254829592f);
    p = p * t;
    float e  = __builtin_amdgcn_exp2f(ax * ax * NLOG2E);   // exp(-x^2)
    float r  = __builtin_fmaf(-p, e, 1.0f);                // underflow -> 1 OK
    return __builtin_copysignf(r, x);
}

__device__ __forceinline__ float gelu_fast(float v) {
    // 0.5*v*(1+erf(v/sqrt(2))) = v*(0.5 + 0.5*erf(v*0.7071...))
    float e = erf_fast(v * 0.70710678118654752f);
    return v * __builtin_fmaf(0.5f, e, 0.5f);
}

__device__ __forceinline__ v8f wmma_f32_4(v2f a, v2f b, v8f c) {
    return __builtin_amdgcn_wmma_f32_16x16x4_f32(
        /*neg_a=*/false, a, /*neg_b=*/false, b,
        /*c_mod=*/(short)0, c, /*reuse_a=*/false, /*reuse_b=*/false);
}

__launch_bounds__(256, 4)
__global__ void mlp_gridsample_wmma_kernel(
    const float* __restrict__ x,     // (B,3)
    const float* __restrict__ fm,    // (4,1024,1024)
    const float* __restrict__ W1, const float* __restrict__ b1,   // (16,8),(16)
    const float* __restrict__ W2, const float* __restrict__ b2,   // (16,16),(16)
    const float* __restrict__ W3, const float* __restrict__ b3,   // (16,16),(16)
    const float* __restrict__ Wh, const float* __restrict__ bh,   // (3,16),(3)
    float* __restrict__ out)                                      // (B,3)
{
    __shared__ float s_in[256][9];   // 8 input features + pad (bank-conflict-free)
    __shared__ float s_h [256][17];  // 16 hidden activations + pad

    const int tid   = threadIdx.x;        // 0..255
    const int lane  = tid & 31;
    const int wbase = tid & ~31;          // wave's first point within block
    const int n     = lane & 15;          // WMMA column (B/C layouts)
    const int khalf = lane >> 4;          // lane half selects K pair within chunk
    const int p     = blockIdx.x * 256 + tid;

    // ---------------- Phase 1: per-point input assembly ----------------
    const float x0 = x[p * 3 + 0];
    const float x1 = x[p * 3 + 1];
    const float x2 = x[p * 3 + 2];
    const float texp = expm1f(x2);

    // grid_sample(bilinear, border, align_corners=False) with g = x*2-1:
    // ix = clamp(x0*W - 0.5, 0, W-1), iy = clamp(x1*H - 0.5, 0, H-1)
    float ix = fminf(fmaxf(x0 * (float)FM_W - 0.5f, 0.0f), (float)(FM_W - 1));
    float iy = fminf(fmaxf(x1 * (float)FM_H - 0.5f, 0.0f), (float)(FM_H - 1));
    float xf = floorf(ix), yf = floorf(iy);
    float wx = ix - xf,    wy = iy - yf;
    int xi0 = (int)xf, yi0 = (int)yf;
    int xi1 = min(xi0 + 1, FM_W - 1);
    int yi1 = min(yi0 + 1, FM_H - 1);
    int o00 = yi0 * FM_W + xi0, o01 = yi0 * FM_W + xi1;
    int o10 = yi1 * FM_W + xi0, o11 = yi1 * FM_W + xi1;

    float feat[4];
#pragma unroll
    for (int c = 0; c < 4; ++c) {
        const float* f = fm + c * FM_HW;   // gathers: 16 MB map resides in L2
        float top = f[o00] * (1.0f - wx) + f[o01] * wx;
        float bot = f[o10] * (1.0f - wx) + f[o11] * wx;
        feat[c] = top * (1.0f - wy) + bot * wy;
    }

    s_in[tid][0] = x0;      s_in[tid][1] = x1;      s_in[tid][2] = x2;
    s_in[tid][3] = feat[0]; s_in[tid][4] = feat[1]; s_in[tid][5] = feat[2];
    s_in[tid][6] = feat[3]; s_in[tid][7] = texp;
    __syncthreads();

    // ---------------- Layer 1: (16x8)@(8x16), 2 K-chunks, 2 tiles ----------------
    v8f c0, c1;
    {
        const float bv = b1[n];
#pragma unroll
        for (int i = 0; i < 8; ++i) { c0[i] = bv; c1[i] = bv; }
    }
#pragma unroll
    for (int c = 0; c < 2; ++c) {
        const int k0 = 4 * c + 2 * khalf;
        v2f bm; bm.x = W1[n * 8 + k0]; bm.y = W1[n * 8 + k0 + 1];
        v2f a0; a0.x = s_in[wbase + n][k0];      a0.y = s_in[wbase + n][k0 + 1];
        v2f a1; a1.x = s_in[wbase + 16 + n][k0]; a1.y = s_in[wbase + 16 + n][k0 + 1];
        c0 = wmma_f32_4(a0, bm, c0);
        c1 = wmma_f32_4(a1, bm, c1);
    }
    // GELU in C layout (VGPR r: row = r + 8*khalf, col = n), scatter to LDS
#pragma unroll
    for (int r = 0; r < 8; ++r) {
        s_h[wbase + r + 8 * khalf][n]      = gelu_fast(c0[r]);
        s_h[wbase + 16 + r + 8 * khalf][n] = gelu_fast(c1[r]);
    }
    __syncthreads();

    // ---------------- Layers 2 & 3: (16x16)@(16x16), 4 K-chunks, 2 tiles -------
#pragma unroll
    for (int layer = 0; layer < 2; ++layer) {
        const float* Wt = (layer == 0) ? W2 : W3;
        const float* bt = (layer == 0) ? b2 : b3;
        v8f d0, d1;
        {
            const float bv = bt[n];
#pragma unroll
            for (int i = 0; i < 8; ++i) { d0[i] = bv; d1[i] = bv; }
        }
#pragma unroll
        for (int c = 0; c < 4; ++c) {
            const int k0 = 4 * c + 2 * khalf;
            v2f bm; bm.x = Wt[n * 16 + k0]; bm.y = Wt[n * 16 + k0 + 1];
            v2f a0; a0.x = s_h[wbase + n][k0];      a0.y = s_h[wbase + n][k0 + 1];
            v2f a1; a1.x = s_h[wbase + 16 + n][k0]; a1.y = s_h[wbase + 16 + n][k0 + 1];
            d0 = wmma_f32_4(a0, bm, d0);
            d1 = wmma_f32_4(a1, bm, d1);
        }
        __syncthreads();   // all reads of previous activations done before overwrite
#pragma unroll
        for (int r = 0; r < 8; ++r) {
            s_h[wbase + r + 8 * khalf][n]      = gelu_fast(d0[r]);
            s_h[wbase + 16 + r + 8 * khalf][n] = gelu_fast(d1[r]);
        }
        __syncthreads();
    }

    // ---------------- Head: (16->3) in VALU, Wh/bh are wave-uniform -------------
    float a0 = bh[0], a1 = bh[1], a2 = bh[2];
#pragma unroll
    for (int k = 0; k < 16; ++k) {
        const float hv = s_h[tid][k];
        a0 += hv * Wh[0 * 16 + k];
        a1 += hv * Wh[1 * 16 + k];
        a2 += hv * Wh[2 * 16 + k];
    }
    out[p * 3 + 0] = a0;
    out[p * 3 + 1] = a1;
    out[p * 3 + 2] = a2;
}

extern "C" void kernel_launch(void* const* d_in, const int* in_sizes, int n_in,
                              void* d_out, int out_size, void* d_ws, size_t ws_size,
                              hipStream_t stream) {
    const float* x  = (const float*)d_in[0];
    const float* fm = (const float*)d_in[1];
    const float* W1 = (const float*)d_in[2];
    const float* b1 = (const float*)d_in[3];
    const float* W2 = (const float*)d_in[4];
    const float* b2 = (const float*)d_in[5];
    const float* W3 = (const float*)d_in[6];
    const float* b3 = (const float*)d_in[7];
    const float* Wh = (const float*)d_in[8];
    const float* bh = (const float*)d_in[9];
    float* out = (float*)d_out;

    const int B = in_sizes[0] / 3;        // 4,194,304 (multiple of 256)
    const int blocks = B / 256;

    mlp_gridsample_wmma_kernel<<<blocks, 256, 0, stream>>>(
        x, fm, W1, b1, W2, b2, W3, b3, Wh, bh, out);
}